// Net_7086696038552
// MI455X (gfx1250) — compile-verified
//
#include <hip/hip_runtime.h>
#include <hip/hip_bf16.h>
#include <math.h>

typedef float v2f __attribute__((ext_vector_type(2)));
typedef float v8f __attribute__((ext_vector_type(8)));

// ---------------------------------------------------------------------------
// deg init over the PADDED node range: every node has a self-loop -> deg starts at 1.0.
// Padding nodes also get 1.0 so rsqrt stays finite.
__global__ void init_deg(float* __restrict__ deg, int np) {
    int i = blockIdx.x * blockDim.x + threadIdx.x;
    if (i < np) deg[i] = 1.0f;
}

// deg count over target index (col)
__global__ void deg_count(const long long* __restrict__ cols, float* __restrict__ deg, int E) {
    int e = blockIdx.x * blockDim.x + threadIdx.x;
    if (e >= E) return;
    atomicAdd(&deg[(int)cols[e]], 1.0f);
}

// dinv = rsqrt(deg) in place (deg >= 1 always due to self-loops)
__global__ void deg_to_dinv(float* __restrict__ deg, int np) {
    int i = blockIdx.x * blockDim.x + threadIdx.x;
    if (i < np) deg[i] = rsqrtf(deg[i]);
}

// ---------------------------------------------------------------------------
// Layer-1 transform via WMMA f32 16x16x4:
//   hs1[node][0:16] = dinv[node]*(x[node][0:3] @ W1[3][16])
// One wave handles 16 nodes. A: M=node(16) x K=4 (K=3 padded); B: K=4 x N=16.
// Node dimension of hs1 is padded to a multiple of 16 -> all loads/stores unconditional.
__global__ void __launch_bounds__(256)
wmma_transform1(const float* __restrict__ x, const float* __restrict__ W1,
                const float* __restrict__ dinv, float* __restrict__ hs1, int n) {
    int lane = threadIdx.x & 31;
    int wave = (blockIdx.x * blockDim.x + threadIdx.x) >> 5;
    int base = wave * 16;
    if (base >= n) return;               // wave-uniform exit; EXEC full otherwise

    int m = lane & 15;                   // M row (node) for A; N col (feat) for B/D
    int h = lane >> 4;                   // half-wave select
    int node = base + m;
    int nodec = node < n ? node : n - 1; // clamp: garbage only reaches padding rows
    float di = dinv[nodec];

    // A operand: lane holds K = 2h, 2h+1 of row M=m (K=3, padded with 0 at K=3)
    int k0 = 2 * h;                                   // 0 or 2 (< 3 always)
    float a0 = x[nodec * 3 + k0] * di;
    float a1 = (k0 + 1 < 3) ? x[nodec * 3 + k0 + 1] * di : 0.0f;

    // B operand: lane holds K = 2h, 2h+1 of column N=m. W1 row-major [k*16 + n]
    float b0 = W1[k0 * 16 + m];
    float b1v = (k0 + 1 < 3) ? W1[(k0 + 1) * 16 + m] : 0.0f;

    v2f a = {a0, a1};
    v2f b = {b0, b1v};
    v8f c = {};
    c = __builtin_amdgcn_wmma_f32_16x16x4_f32(false, a, false, b, (short)0, c, false, false);

    // D: lane holds feat N=m for nodes M = v + 8h. 16 lanes per half-wave write one
    // contiguous 64B row per v. Unconditional: padding rows land in workspace padding.
    #pragma unroll
    for (int v = 0; v < 8; ++v) {
        int nd = base + v + 8 * h;
        hs1[nd * 16 + m] = c[v];
    }
}

// ---------------------------------------------------------------------------
// Fused finalize1 + layer-2 transform:
//   out1[node][k]  = relu(b1[k] + dinv*(acc1 + hs1))          (self-loop folded in)
//   hs2[node][0:7] = dinv[node] * (out1[node] @ W2[16][7])    (4 chained 16x16x4 WMMAs, N padded to 16)
__global__ void __launch_bounds__(256)
wmma_transform2(const float* __restrict__ acc1, const float* __restrict__ hs1,
                const float* __restrict__ dinv, const float* __restrict__ bias1,
                const float* __restrict__ W2, float* __restrict__ hs2, int n) {
    int lane = threadIdx.x & 31;
    int wave = (blockIdx.x * blockDim.x + threadIdx.x) >> 5;
    int base = wave * 16;
    if (base >= n) return;

    int m = lane & 15;
    int h = lane >> 4;
    int node = base + m;
    int nodec = node < n ? node : n - 1;
    float di = dinv[nodec];

    v8f c = {};
    #pragma unroll
    for (int kk = 0; kk < 4; ++kk) {
        int k0 = 4 * kk + 2 * h;         // K pair handled by this lane in this chunk
        // acc1/hs1 reads are on clamped (valid, padded) rows -> unconditional float2 loads
        float t0 = bias1[k0]     + di * (acc1[nodec * 16 + k0]     + hs1[nodec * 16 + k0]);
        float t1 = bias1[k0 + 1] + di * (acc1[nodec * 16 + k0 + 1] + hs1[nodec * 16 + k0 + 1]);
        float a0 = di * fmaxf(t0, 0.0f); // prescale layer-2 input by dinv[node]
        float a1 = di * fmaxf(t1, 0.0f);
        // B: W2 is [16][7] row-major (k*7 + n); columns 7..15 padded with 0
        float bb0 = (m < 7) ? W2[k0 * 7 + m] : 0.0f;
        float bb1 = (m < 7) ? W2[(k0 + 1) * 7 + m] : 0.0f;
        v2f a = {a0, a1};
        v2f b = {bb0, bb1};
        c = __builtin_amdgcn_wmma_f32_16x16x4_f32(false, a, false, b, (short)0, c, false, false);
    }

    // store feats 0..7 (col 7 is exact zero from padded B); hs2 stride 8, padded node dim
    if (m < 8) {
        #pragma unroll
        for (int v = 0; v < 8; ++v) {
            int nd = base + v + 8 * h;
            hs2[nd * 8 + m] = c[v];
        }
    }
}

// ---------------------------------------------------------------------------
// Edge scatter, layer 1: acc1[col][0:16] += hs1[row][0:16]
// (norm pre-folded into hs1; dinv[col] applied at finalize -> zero per-edge math)
__global__ void scatter16(const long long* __restrict__ rows, const long long* __restrict__ cols,
                          const float* __restrict__ hs, float* __restrict__ acc, int E) {
    int e = blockIdx.x * blockDim.x + threadIdx.x;
    if (e >= E) return;
    int r = (int)rows[e];
    int c = (int)cols[e];
    const float4* src = (const float4*)(hs + (size_t)r * 16);
    float4 v0 = src[0], v1 = src[1], v2 = src[2], v3 = src[3];
    float* dst = acc + (size_t)c * 16;
    atomicAdd(dst + 0,  v0.x); atomicAdd(dst + 1,  v0.y);
    atomicAdd(dst + 2,  v0.z); atomicAdd(dst + 3,  v0.w);
    atomicAdd(dst + 4,  v1.x); atomicAdd(dst + 5,  v1.y);
    atomicAdd(dst + 6,  v1.z); atomicAdd(dst + 7,  v1.w);
    atomicAdd(dst + 8,  v2.x); atomicAdd(dst + 9,  v2.y);
    atomicAdd(dst + 10, v2.z); atomicAdd(dst + 11, v2.w);
    atomicAdd(dst + 12, v3.x); atomicAdd(dst + 13, v3.y);
    atomicAdd(dst + 14, v3.z); atomicAdd(dst + 15, v3.w);
}

// Edge scatter, layer 2: 7 features (stride-8 storage)
__global__ void scatter7(const long long* __restrict__ rows, const long long* __restrict__ cols,
                         const float* __restrict__ hs, float* __restrict__ acc, int E) {
    int e = blockIdx.x * blockDim.x + threadIdx.x;
    if (e >= E) return;
    int r = (int)rows[e];
    int c = (int)cols[e];
    const float4* src = (const float4*)(hs + (size_t)r * 8);
    float4 v0 = src[0], v1 = src[1];
    float* dst = acc + (size_t)c * 8;
    atomicAdd(dst + 0, v0.x); atomicAdd(dst + 1, v0.y);
    atomicAdd(dst + 2, v0.z); atomicAdd(dst + 3, v0.w);
    atomicAdd(dst + 4, v1.x); atomicAdd(dst + 5, v1.y);
    atomicAdd(dst + 6, v1.z);
}

// ---------------------------------------------------------------------------
// Finalize layer 2 (+ self-loop term) and log_softmax over 7 classes
__global__ void finalize(const float* __restrict__ acc2, const float* __restrict__ hs2,
                         const float* __restrict__ dinv, const float* __restrict__ bias2,
                         float* __restrict__ out, int n) {
    int i = blockIdx.x * blockDim.x + threadIdx.x;
    if (i >= n) return;
    float di = dinv[i];
    float v[7];
    float mx = -INFINITY;
    #pragma unroll
    for (int j = 0; j < 7; ++j) {
        v[j] = bias2[j] + di * (acc2[i * 8 + j] + hs2[i * 8 + j]);
        mx = fmaxf(mx, v[j]);
    }
    float s = 0.0f;
    #pragma unroll
    for (int j = 0; j < 7; ++j) s += expf(v[j] - mx);
    float lse = logf(s);
    #pragma unroll
    for (int j = 0; j < 7; ++j) out[i * 7 + j] = v[j] - mx - lse;
}

// ---------------------------------------------------------------------------
extern "C" void kernel_launch(void* const* d_in, const int* in_sizes, int n_in,
                              void* d_out, int out_size, void* d_ws, size_t ws_size,
                              hipStream_t stream) {
    const float*     x    = (const float*)d_in[0];
    const long long* ei   = (const long long*)d_in[1];   // int64 edge_index [2, E]
    const float*     W1   = (const float*)d_in[2];
    const float*     b1   = (const float*)d_in[3];
    const float*     W2   = (const float*)d_in[4];
    const float*     b2   = (const float*)d_in[5];
    float* out = (float*)d_out;

    const int N  = in_sizes[0] / 3;
    const int Np = (N + 15) & ~15;       // node dim padded to WMMA tile (16)
    const int E  = in_sizes[1] / 2;
    const long long* rows = ei;          // source
    const long long* cols = ei + E;      // target

    // Workspace layout (floats), all float4-aligned (Np multiple of 16).
    float* ws   = (float*)d_ws;
    float* dinv = ws;                         // [Np]     deg -> dinv in place
    float* hs1  = ws + (size_t)Np;            // [Np*16]  dinv * (x @ W1)
    float* acc1 = ws + (size_t)Np * 17;       // [Np*16]  scatter accumulator (no self-loop)
    float* hs2  = ws + (size_t)Np * 33;       // [Np*8]   dinv * (relu-out1 @ W2), feat 7 = 0
    float* acc2 = ws + (size_t)Np * 41;       // [Np*8]

    const int B = 256;
    int nodeBlocks  = (N + B - 1) / B;
    int padBlocks   = (Np + B - 1) / B;
    int edgeBlocks  = (E + B - 1) / B;
    int waveThreads = (Np / 16) * 32;         // one wave per 16 nodes
    int waveBlocks  = (waveThreads + B - 1) / B;

    // zero accumulators (graph-capture safe)
    hipMemsetAsync(acc1, 0, (size_t)Np * 16 * sizeof(float), stream);
    hipMemsetAsync(acc2, 0, (size_t)Np * 8 * sizeof(float), stream);

    // degree / normalization
    init_deg<<<padBlocks, B, 0, stream>>>(dinv, Np);
    deg_count<<<edgeBlocks, B, 0, stream>>>(cols, dinv, E);
    deg_to_dinv<<<padBlocks, B, 0, stream>>>(dinv, Np);

    // layer 1
    wmma_transform1<<<waveBlocks, B, 0, stream>>>(x, W1, dinv, hs1, N);
    scatter16<<<edgeBlocks, B, 0, stream>>>(rows, cols, hs1, acc1, E);

    // layer 2 (bias+relu+prescale fused into WMMA A-operand construction)
    wmma_transform2<<<waveBlocks, B, 0, stream>>>(acc1, hs1, dinv, b1, W2, hs2, N);
    scatter7<<<edgeBlocks, B, 0, stream>>>(rows, cols, hs2, acc2, E);

    // finalize + log_softmax
    finalize<<<nodeBlocks, B, 0, stream>>>(acc2, hs2, dinv, b2, out, N);
}